// Bigram_24618752540964
// MI455X (gfx1250) — compile-verified
//
#include <hip/hip_runtime.h>
#include <hip/hip_bf16.h>
#include <math.h>

#define VOCAB 8192
#define BLOCK 256
#define WAVES (BLOCK / 32)
#define CHUNKS 8                    // 8 x float4 = 32 logits per thread

typedef float v4f  __attribute__((ext_vector_type(4)));
typedef unsigned int u32x4 __attribute__((ext_vector_type(4)));
typedef int i32x4 __attribute__((ext_vector_type(4)));
typedef int i32x8 __attribute__((ext_vector_type(8)));

// -----------------------------------------------------------------------------
// Kernel 1: per-token fused gather + bias + log-softmax + NT logits store.
// grid = B*T workgroups, 256 threads (8 wave32s).
// -----------------------------------------------------------------------------
__global__ __launch_bounds__(BLOCK) void bigram_logits_ce(
    const int* __restrict__ idx, const int* __restrict__ targets,
    const float* __restrict__ W, const float* __restrict__ b,
    float* __restrict__ logits, float* __restrict__ nll_ws)
{
    __shared__ float sh_b[VOCAB];       // 32 KB staged bias
    __shared__ float red_max[WAVES];
    __shared__ float red_sum[WAVES];

    const int tid   = threadIdx.x;
    const int lane  = tid & 31;
    const int wid   = tid >> 5;
    const int token = blockIdx.x;

    // ---- Stage bias vector into LDS via the Tensor Data Mover (wave 0 only;
    // ---- TDM ignores EXEC, so gate by wave, not lane) ----
#if defined(__HIP_DEVICE_COMPILE__) && defined(__gfx1250__)
    if (wid == 0) {
        const unsigned lds_off = (unsigned)(size_t)(&sh_b[0]);
        const unsigned long long ga = (unsigned long long)(size_t)b;
        u32x4 g0;
        g0[0] = 1u;                                          // count=1 (valid D#)
        g0[1] = lds_off;                                     // lds_addr
        g0[2] = (unsigned)(ga & 0xFFFFFFFFull);              // global_addr lo
        g0[3] = (unsigned)((ga >> 32) & 0x01FFFFFFull)       // global_addr hi
              | (2u << 30);                                  // type=2 ("image")
        i32x8 g1;
        g1[0] = (int)(2u << 16);        // data_size=2 -> 4 bytes; wg_mask=0
        g1[1] = (int)(0x2000u << 16);   // tensor_dim0 = 8192 (bits 79:48 lo half)
        g1[2] = (int)(1u << 16);        // tensor_dim0 hi = 0; tensor_dim1 = 1
        g1[3] = (int)(0x2000u << 16);   // tensor_dim1 hi = 0; tile_dim0 = 8192
        g1[4] = 1;                      // tile_dim1 = 1; tile_dim2 = 0
        g1[5] = 8192;                   // tensor_dim0_stride lo
        g1[6] = 0;                      // stride hi / dim1_stride lo
        g1[7] = 0;
        i32x4 z4 = {0, 0, 0, 0};
#if __clang_major__ >= 23
        i32x8 z8 = {0, 0, 0, 0, 0, 0, 0, 0};
        __builtin_amdgcn_tensor_load_to_lds(g0, g1, z4, z4, z8, 0);
#else
        __builtin_amdgcn_tensor_load_to_lds(g0, g1, z4, z4, 0);
#endif
        __builtin_amdgcn_s_wait_tensorcnt(0);
    }
#else
    for (int j = tid; j < VOCAB; j += BLOCK) sh_b[j] = b[j];
#endif

    const int iv = idx[token];          // gathered column of W
    __syncthreads();                    // bias visible to all 8 waves

    // ---- Pass A: gather W[:, iv] + bias, keep in VGPRs, track max ----
    v4f vals[CHUNKS];
    float m = -INFINITY;
#pragma unroll
    for (int k = 0; k < CHUNKS; ++k) {
        const int j = k * (BLOCK * 4) + tid * 4;
        v4f bb = *(const v4f*)(&sh_b[j]);           // ds_load_b128
        v4f w;
        w.x = W[(size_t)(j + 0) * VOCAB + iv];      // strided column gather,
        w.y = W[(size_t)(j + 1) * VOCAB + iv];      // RT-cached (L2 holds W)
        w.z = W[(size_t)(j + 2) * VOCAB + iv];
        w.w = W[(size_t)(j + 3) * VOCAB + iv];
        v4f v = w + bb;
        vals[k] = v;
        m = fmaxf(m, fmaxf(fmaxf(v.x, v.y), fmaxf(v.z, v.w)));
    }

    // ---- block max (wave32 shfl + LDS across 8 waves) ----
#pragma unroll
    for (int off = 16; off > 0; off >>= 1)
        m = fmaxf(m, __shfl_xor(m, off, 32));
    if (lane == 0) red_max[wid] = m;
    __syncthreads();
    float M = red_max[0];
#pragma unroll
    for (int wv = 1; wv < WAVES; ++wv) M = fmaxf(M, red_max[wv]);

    // ---- Pass B: sum exp(v - M); stream logits out non-temporally ----
    float* out_row = logits + (size_t)token * VOCAB;
    float s = 0.f;
#pragma unroll
    for (int k = 0; k < CHUNKS; ++k) {
        const int j = k * (BLOCK * 4) + tid * 4;
        v4f v = vals[k];
        s += __expf(v.x - M) + __expf(v.y - M)
           + __expf(v.z - M) + __expf(v.w - M);
        __builtin_nontemporal_store(v, (v4f*)(out_row + j));  // b128 NT store
    }
#pragma unroll
    for (int off = 16; off > 0; off >>= 1)
        s += __shfl_xor(s, off, 32);
    if (lane == 0) red_sum[wid] = s;
    __syncthreads();

    if (tid == 0) {
        float S = 0.f;
        for (int wv = 0; wv < WAVES; ++wv) S += red_sum[wv];
        const int t = targets[token];
        float nll = 0.f;
        if (t >= 0) {
            const float vt = W[(size_t)t * VOCAB + iv] + sh_b[t];
            nll = (M + __logf(S)) - vt;     // logZ - logit[target]
        }
        nll_ws[token] = nll;                // 0 for ignored tokens
    }
}

// -----------------------------------------------------------------------------
// Kernel 2: reduce 16384 per-token NLLs -> mean loss over valid tokens.
// Single workgroup, deterministic order, no atomics / no persistent state.
// -----------------------------------------------------------------------------
__global__ __launch_bounds__(BLOCK) void bigram_loss_reduce(
    const float* __restrict__ nll_ws, const int* __restrict__ targets,
    float* __restrict__ loss_out, int n)
{
    __shared__ float s_s[BLOCK];
    __shared__ float s_c[BLOCK];
    const int tid = threadIdx.x;
    float s = 0.f, c = 0.f;
    for (int i = tid; i < n; i += BLOCK) {
        s += nll_ws[i];
        c += (targets[i] != -1) ? 1.f : 0.f;
    }
    s_s[tid] = s;
    s_c[tid] = c;
    __syncthreads();
    for (int off = BLOCK / 2; off > 0; off >>= 1) {
        if (tid < off) {
            s_s[tid] += s_s[tid + off];
            s_c[tid] += s_c[tid + off];
        }
        __syncthreads();
    }
    if (tid == 0) loss_out[0] = s_s[0] / fmaxf(s_c[0], 1.f);
}

// -----------------------------------------------------------------------------
// Launch: inputs in setup_inputs() order: idx, targets, W, b.
// d_out = [B*T*V logits][1 loss], all float32.
// -----------------------------------------------------------------------------
extern "C" void kernel_launch(void* const* d_in, const int* in_sizes, int n_in,
                              void* d_out, int out_size, void* d_ws, size_t ws_size,
                              hipStream_t stream)
{
    const int*   idx = (const int*)d_in[0];
    const int*   tgt = (const int*)d_in[1];
    const float* W   = (const float*)d_in[2];
    const float* b   = (const float*)d_in[3];
    float* out = (float*)d_out;
    float* nll_ws = (float*)d_ws;                 // 16384 floats of scratch

    const int n_tok = in_sizes[0];                // B*T = 16384

    bigram_logits_ce<<<dim3((unsigned)n_tok), dim3(BLOCK), 0, stream>>>(
        idx, tgt, W, b, out, nll_ws);

    bigram_loss_reduce<<<dim3(1), dim3(BLOCK), 0, stream>>>(
        nll_ws, tgt, out + (size_t)n_tok * VOCAB, n_tok);
}